// Structure2vec_1666447311066
// MI455X (gfx1250) — compile-verified
//
#include <hip/hip_runtime.h>
#include <cmath>

typedef __attribute__((ext_vector_type(2))) float v2f;
typedef __attribute__((ext_vector_type(8))) float v8f;

#define N_NODES 100000
#define N_EDGES 3200000
#define IN_DIM 128
#define OUT_DIM 64
#define NTILES (N_NODES / 16)   // 6250 exact

// Pair-interleaved LDS weight layout:
//   sW[((k>>1)*64 + n)*2 + (k&1)] = W[k][n]
// so the B fragment (W[k][n], W[k+1][n]) for even k is one aligned b64 load.
__device__ __forceinline__ void stage_weights_interleaved(
    float* __restrict__ sW, const float* __restrict__ W, int kdim) {
  for (int i = threadIdx.x; i < kdim * OUT_DIM; i += 256) {
    const int k = i >> 6;          // / OUT_DIM
    const int n = i & 63;          // % OUT_DIM
    sW[(((k >> 1) << 6) + n) * 2 + (k & 1)] = W[i];
  }
}

// ---------------------------------------------------------------------------
// K1: base = feature @ W_lin  (WMMA f32 16x16x4), u1 = tanh(base), zero m-tile
// One wave computes a 16x64 output tile; K=128 in steps of 4 (32 steps x 4 Ntiles)
// ---------------------------------------------------------------------------
__global__ __launch_bounds__(256) void k_base(const float* __restrict__ feature,
                                              const float* __restrict__ Wlin,
                                              float* __restrict__ base,  // = d_out
                                              float* __restrict__ u1,
                                              float* __restrict__ m) {
  __shared__ float sW[IN_DIM * OUT_DIM];  // 32 KB, pair-interleaved
  stage_weights_interleaved(sW, Wlin, IN_DIM);
  __syncthreads();

  const int wave = threadIdx.x >> 5;
  const int lane = threadIdx.x & 31;
  const int tile = blockIdx.x * 8 + wave;
  if (tile >= NTILES) return;

  const int hi   = lane >> 4;   // lane half selects K pair
  const int mrow = lane & 15;   // row within tile (A) / col within N-tile (B,C)

  v8f acc[4] = {v8f{}, v8f{}, v8f{}, v8f{}};
  const float* arow = feature + (size_t)(tile * 16 + mrow) * IN_DIM;

#pragma unroll
  for (int kk = 0; kk < IN_DIM; kk += 4) {
    // A frag: lanes 0-15 hold K=kk,kk+1 ; lanes 16-31 hold K=kk+2,kk+3
    v2f a = *(const v2f*)(arow + kk + 2 * hi);
    const int p = (kk >> 1) + hi;                 // K-pair index for this lane half
    const float* bp = sW + (((p << 6) + mrow) << 1);
#pragma unroll
    for (int j = 0; j < 4; ++j) {
      v2f b = *(const v2f*)(bp + (j << 5));       // (j*16 cols) * 2 interleave
      acc[j] = __builtin_amdgcn_wmma_f32_16x16x4_f32(false, a, false, b,
                                                     (short)0, acc[j], false, false);
    }
  }

  // zero this tile's rows of m (16x64 contiguous floats), float4-coalesced
  float4* mt = (float4*)(m + (size_t)tile * 16 * OUT_DIM);
  const float4 z4 = {0.f, 0.f, 0.f, 0.f};
#pragma unroll
  for (int i = 0; i < 8; ++i) mt[i * 32 + lane] = z4;

  // C layout: vgpr r -> row r + 8*hi, col = j*16 + mrow
#pragma unroll
  for (int j = 0; j < 4; ++j) {
#pragma unroll
    for (int r = 0; r < 8; ++r) {
      const int node = tile * 16 + r + 8 * hi;
      const int col  = j * 16 + mrow;
      const float v  = acc[j][r];
      base[(size_t)node * OUT_DIM + col] = v;
      u1[(size_t)node * OUT_DIM + col]   = tanhf(v);
    }
  }
}

// ---------------------------------------------------------------------------
// K2: m[dst[e]] += u1[src[e]]  — one wave per edge, lane handles 2 columns.
// u1/m (25.6 MB each) stay L2-resident; the once-streamed index arrays are
// loaded non-temporally so they don't evict the working set.
// ---------------------------------------------------------------------------
__global__ __launch_bounds__(256) void k_scatter(const int* __restrict__ src,
                                                 const int* __restrict__ dst,
                                                 const float* __restrict__ u1,
                                                 float* __restrict__ m) {
  const long long tid = (long long)blockIdx.x * blockDim.x + threadIdx.x;
  const long long e   = tid >> 5;
  if (e >= (long long)N_EDGES) return;
  const int lane = (int)(tid & 31);
  const int s = __builtin_nontemporal_load(src + e);
  const int d = __builtin_nontemporal_load(dst + e);
  const v2f v = *(const v2f*)(u1 + (size_t)s * OUT_DIM + lane * 2);
  float* p = m + (size_t)d * OUT_DIM + lane * 2;
  atomicAdd(p, v.x);
  atomicAdd(p + 1, v.y);
}

// ---------------------------------------------------------------------------
// K3: out = tanh(base + relu(m @ W_d1) @ W_d2) — two WMMA GEMMs per tile,
// with a per-wave LDS round trip to reshape C-layout -> A-layout between them.
// base lives in d_out and is overwritten in place.
// ---------------------------------------------------------------------------
__global__ __launch_bounds__(256) void k_dense(const float* __restrict__ m,
                                               const float* __restrict__ Wd1,
                                               const float* __restrict__ Wd2,
                                               float* __restrict__ out) {
  __shared__ float sW1[OUT_DIM * OUT_DIM];   // 16 KB, pair-interleaved
  __shared__ float sW2[OUT_DIM * OUT_DIM];   // 16 KB, pair-interleaved
  __shared__ float sT[8][16 * OUT_DIM];      // 32 KB (4 KB per wave)
  stage_weights_interleaved(sW1, Wd1, OUT_DIM);
  stage_weights_interleaved(sW2, Wd2, OUT_DIM);
  __syncthreads();

  const int wave = threadIdx.x >> 5;
  const int lane = threadIdx.x & 31;
  const int tile = blockIdx.x * 8 + wave;
  if (tile >= NTILES) return;

  const int hi   = lane >> 4;
  const int mrow = lane & 15;

  // GEMM 1: t = m_tile @ W_d1
  v8f acc[4] = {v8f{}, v8f{}, v8f{}, v8f{}};
  const float* arow = m + (size_t)(tile * 16 + mrow) * OUT_DIM;
#pragma unroll
  for (int kk = 0; kk < OUT_DIM; kk += 4) {
    v2f a = *(const v2f*)(arow + kk + 2 * hi);
    const int p = (kk >> 1) + hi;
    const float* bp = sW1 + (((p << 6) + mrow) << 1);
#pragma unroll
    for (int j = 0; j < 4; ++j) {
      v2f b = *(const v2f*)(bp + (j << 5));
      acc[j] = __builtin_amdgcn_wmma_f32_16x16x4_f32(false, a, false, b,
                                                     (short)0, acc[j], false, false);
    }
  }

  // relu + stage into this wave's LDS region (C-layout -> row-major)
  float* t = sT[wave];
#pragma unroll
  for (int j = 0; j < 4; ++j) {
#pragma unroll
    for (int r = 0; r < 8; ++r) {
      t[(r + 8 * hi) * OUT_DIM + j * 16 + mrow] = fmaxf(acc[j][r], 0.f);
    }
  }
  // wave32 lanes run in lockstep; drain DS writes before cross-lane reads
  asm volatile("s_wait_dscnt 0" ::: "memory");

  // GEMM 2: h = t @ W_d2  (A frags re-read from LDS in A-layout)
  v8f acc2[4] = {v8f{}, v8f{}, v8f{}, v8f{}};
  const float* trow = t + mrow * OUT_DIM;
#pragma unroll
  for (int kk = 0; kk < OUT_DIM; kk += 4) {
    v2f a = *(const v2f*)(trow + kk + 2 * hi);
    const int p = (kk >> 1) + hi;
    const float* bp = sW2 + (((p << 6) + mrow) << 1);
#pragma unroll
    for (int j = 0; j < 4; ++j) {
      v2f b = *(const v2f*)(bp + (j << 5));
      acc2[j] = __builtin_amdgcn_wmma_f32_16x16x4_f32(false, a, false, b,
                                                      (short)0, acc2[j], false, false);
    }
  }

  // out = tanh(base + h); base was staged in d_out by k_base
#pragma unroll
  for (int j = 0; j < 4; ++j) {
#pragma unroll
    for (int r = 0; r < 8; ++r) {
      const size_t idx = (size_t)(tile * 16 + r + 8 * hi) * OUT_DIM + j * 16 + mrow;
      out[idx] = tanhf(out[idx] + acc2[j][r]);
    }
  }
}

// ---------------------------------------------------------------------------
extern "C" void kernel_launch(void* const* d_in, const int* in_sizes, int n_in,
                              void* d_out, int out_size, void* d_ws, size_t ws_size,
                              hipStream_t stream) {
  const float* feature = (const float*)d_in[0];
  const int*   src     = (const int*)d_in[1];
  const int*   dst     = (const int*)d_in[2];
  const float* Wlin    = (const float*)d_in[3];
  const float* Wd1     = (const float*)d_in[4];
  const float* Wd2     = (const float*)d_in[5];
  float* out = (float*)d_out;

  float* u1 = (float*)d_ws;                       // 25.6 MB
  float* m  = u1 + (size_t)N_NODES * OUT_DIM;     // 25.6 MB

  const dim3 blk(256);
  const int tileBlocks = (NTILES + 7) / 8;

  k_base<<<tileBlocks, blk, 0, stream>>>(feature, Wlin, out, u1, m);

  const long long sthreads = (long long)N_EDGES * 32;
  const int sblocks = (int)((sthreads + 255) / 256);
  k_scatter<<<sblocks, blk, 0, stream>>>(src, dst, u1, m);

  k_dense<<<tileBlocks, blk, 0, stream>>>(m, Wd1, Wd2, out);
}